// GraphAttentionLayer_6854767805053
// MI455X (gfx1250) — compile-verified
//
#include <hip/hip_runtime.h>

#define NODES  50000
#define EDGES  640000
#define INDIM  128
#define OUTDIM 128
#define HEADS  4
#define DHEAD  32
#define RUSED  4   // relations surviving the [:, :OUT] truncation (head-0 cols only)

typedef float v2f __attribute__((ext_vector_type(2)));
typedef float v8f __attribute__((ext_vector_type(8)));

// ---------------------------------------------------------------------------
// Phase 0: zero the accumulator (d_out) and per-(node,relation) degree counts
// ---------------------------------------------------------------------------
__global__ void rgat_zero(float* __restrict__ agg, float* __restrict__ deg) {
  int t = blockIdx.x * blockDim.x + threadIdx.x;
  if (t < NODES * OUTDIM) agg[t] = 0.0f;
  if (t < NODES * RUSED)  deg[t] = 0.0f;
}

// ---------------------------------------------------------------------------
// Phase 1: y[n, r*32+dh] = sum_k x[n,k] * Ws[r,k,dh] + bs[r,dh],  r=0..3,dh=0..31
// One wave per 16-node tile; A panel (16x128) held in registers (32 x v2f),
// loop over 8 column tiles; 32 x V_WMMA_F32_16X16X4_F32 per tile.
// ---------------------------------------------------------------------------
__global__ void rgat_gemm(const float* __restrict__ x,
                          const float* __restrict__ Ws,
                          const float* __restrict__ bs,
                          float* __restrict__ y) {
  const int lane = threadIdx.x & 31;
  const int wave = (blockIdx.x * blockDim.x + threadIdx.x) >> 5;
  const int mtiles = NODES / 16;            // 3125 (exact)
  if (wave >= mtiles) return;               // whole-wave exit: EXEC all-1s below

  const int m_base = wave * 16;
  const int arow   = m_base + (lane & 15);
  const int koff   = (lane >> 4) << 1;      // lanes 0-15: K=0,1 ; lanes 16-31: K=2,3

  // Preload A tile: 32 K-steps of 4; each lane holds 2 consecutive K floats.
  v2f a[32];
#pragma unroll
  for (int ks = 0; ks < 32; ++ks) {
    const float* p = x + (size_t)arow * INDIM + ks * 4 + koff;
    a[ks].x = p[0];
    a[ks].y = p[1];
  }

  for (int nt = 0; nt < 8; ++nt) {
    const int col = nt * 16 + (lane & 15);  // packed output column 0..127
    const int r   = col >> 5;               // relation 0..3
    const int dh  = col & 31;               // head-0 feature
    const float* wbase = Ws + (size_t)r * (INDIM * HEADS * DHEAD) + dh;

    v8f c = {};
#pragma unroll
    for (int ks = 0; ks < 32; ++ks) {
      const int k0 = ks * 4 + koff;
      v2f b;
      b.x = wbase[(size_t)k0 * (HEADS * DHEAD)];
      b.y = wbase[(size_t)(k0 + 1) * (HEADS * DHEAD)];
      // D = A(16x4,f32) x B(4x16,f32) + C(16x16,f32)
      c = __builtin_amdgcn_wmma_f32_16x16x4_f32(false, a[ks], false, b,
                                                (short)0, c, false, false);
    }

    const float bias  = bs[r * (HEADS * DHEAD) + dh];
    const int   rbase = m_base + ((lane >> 4) << 3); // lanes>=16 hold M = v+8
#pragma unroll
    for (int v = 0; v < 8; ++v) {
      y[(size_t)(rbase + v) * OUTDIM + col] = c[v] + bias;
    }
  }
}

// ---------------------------------------------------------------------------
// Phase 2: per-edge gather + scatter-add. One wave (32 lanes) per edge:
// lane = dh, relation from edge_types; relation 4 is truncated away -> skip.
// ---------------------------------------------------------------------------
__global__ void rgat_edge(const float* __restrict__ y,
                          const int* __restrict__ edge_index, // [2,E]
                          const int* __restrict__ edge_types, // [E]
                          float* __restrict__ agg,
                          float* __restrict__ deg) {
  const long t = (long)blockIdx.x * blockDim.x + threadIdx.x;
  const int  e = (int)(t >> 5);
  const int  lane = (int)(t & 31);
  if (e >= EDGES) return;

  const int r = edge_types[e];
  if (r >= RUSED) return;                   // relation 4: dropped by [:, :128]

  const int src = edge_index[e];            // row 0
  const int dst = edge_index[EDGES + e];    // row 1

  const float v = y[(size_t)src * OUTDIM + r * DHEAD + lane];
  atomicAdd(&agg[(size_t)dst * OUTDIM + r * DHEAD + lane], v);
  if (lane == 0) atomicAdd(&deg[(size_t)dst * RUSED + r], 1.0f);
}

// ---------------------------------------------------------------------------
// Phase 3: out[n,c] = agg[n,c] / max(deg[n, c/32], 1)
// ---------------------------------------------------------------------------
__global__ void rgat_div(float* __restrict__ agg, const float* __restrict__ deg) {
  const int t = blockIdx.x * blockDim.x + threadIdx.x;
  if (t >= NODES * OUTDIM) return;
  const int n = t >> 7;
  const int r = (t & 127) >> 5;
  agg[t] = agg[t] / fmaxf(deg[n * RUSED + r], 1.0f);
}

// ---------------------------------------------------------------------------
extern "C" void kernel_launch(void* const* d_in, const int* in_sizes, int n_in,
                              void* d_out, int out_size, void* d_ws, size_t ws_size,
                              hipStream_t stream) {
  const float* x  = (const float*)d_in[0];  // [N,128]
  const float* Ws = (const float*)d_in[1];  // [5,128,128]
  const float* bs = (const float*)d_in[2];  // [5,128]
  const int*   ei = (const int*)d_in[3];    // [2,E]
  const int*   et = (const int*)d_in[4];    // [E]

  float* out = (float*)d_out;                         // [N,128] accumulator + result
  float* y   = (float*)d_ws;                          // [N,128] projected features
  float* deg = y + (size_t)NODES * OUTDIM;            // [N,4] degree counts

  {
    int total = NODES * OUTDIM;
    rgat_zero<<<(total + 255) / 256, 256, 0, stream>>>(out, deg);
  }
  {
    int waves = NODES / 16;                           // 3125 tile-waves
    int threads = waves * 32;
    rgat_gemm<<<(threads + 255) / 256, 256, 0, stream>>>(x, Ws, bs, y);
  }
  {
    long threads = (long)EDGES * 32;
    rgat_edge<<<(int)((threads + 255) / 256), 256, 0, stream>>>(y, ei, et, out, deg);
  }
  {
    int total = NODES * OUTDIM;
    rgat_div<<<(total + 255) / 256, 256, 0, stream>>>(out, deg);
  }
}